// MatrixExponential_10823317586614
// MI455X (gfx1250) — compile-verified
//
#include <hip/hip_runtime.h>

// Batched matrix exponential: 2048 x (128x128) fp32.
// Scaling-and-squaring + degree-12 Taylor (Horner), all matmuls on
// V_WMMA_F32_16X16X4_F32 (fp32 matrix pipe), matrices LDS-resident.

typedef __attribute__((ext_vector_type(2))) float v2f;
typedef __attribute__((ext_vector_type(4))) float v4f;
typedef __attribute__((ext_vector_type(8))) float v8f;

#define N128   128
#define LDSTR  132                 // padded LDS row stride (floats): 16B-aligned rows, 4-bank skew
#define MBUF   (N128 * LDSTR)      // floats per matrix buffer
#define LDS_BYTES (3u * MBUF * sizeof(float))

// D = (A @ B) * scl (+ I if addI).  A,B,D are 128x128 in LDS, row stride LDSTR.
// Wave w computes output columns [16w, 16w+16) for all 128 rows:
// B fragment loaded once per K-step, reused by 8 WMMAs (one per 16-row tile).
__device__ __forceinline__ void mm128(const float* __restrict__ A,
                                      const float* __restrict__ B,
                                      float* __restrict__ D,
                                      float scl, bool addI,
                                      int lane, int wave) {
    __syncthreads();                       // previous phase's writes visible
    const int hi = lane >> 4;              // lane half selects K sub-pair (ISA 16x4 layout)
    const int ln = lane & 15;              // M (for A) / N (for B,C,D)
    const int tj = wave;                   // this wave's 16-wide column block

    v8f acc[8];
#pragma unroll
    for (int t = 0; t < 8; ++t) acc[t] = (v8f)0.0f;

#pragma unroll 4
    for (int k0 = 0; k0 < N128; k0 += 4) {
        const int kk = k0 + 2 * hi;
        // B fragment: VGPR0 = B[kk][n], VGPR1 = B[kk+1][n]  (n = tj*16 + ln)
        v2f bf;
        bf.x = B[kk * LDSTR + tj * 16 + ln];
        bf.y = B[(kk + 1) * LDSTR + tj * 16 + ln];
#pragma unroll
        for (int ti = 0; ti < 8; ++ti) {
            // A fragment: VGPR0 = A[m][kk], VGPR1 = A[m][kk+1] (m = ti*16 + ln); b64 load
            v2f af = *(const v2f*)&A[(ti * 16 + ln) * LDSTR + kk];
            acc[ti] = __builtin_amdgcn_wmma_f32_16x16x4_f32(
                false, af, false, bf, (short)0, acc[ti], false, false);
        }
    }
    __syncthreads();                       // all reads of A/B done before D reuse later
    // C/D layout: VGPR r -> M = ti*16 + r + 8*hi, lane ln -> N = tj*16 + ln
#pragma unroll
    for (int ti = 0; ti < 8; ++ti) {
#pragma unroll
        for (int r = 0; r < 8; ++r) {
            const int row = ti * 16 + 8 * hi + r;
            const int col = tj * 16 + ln;
            float v = acc[ti][r] * scl;
            if (addI && row == col) v += 1.0f;
            D[row * LDSTR + col] = v;
        }
    }
    __syncthreads();
}

__global__ __launch_bounds__(256, 1)
void MatrixExponential_expm_kernel(const float* __restrict__ X,
                                   float* __restrict__ Y) {
    extern __shared__ float lds[];
    float* bufA = lds;                 // scaled input As
    float* bufR = lds + MBUF;          // polynomial / result
    float* bufT = lds + 2 * MBUF;      // matmul temp

    __shared__ float rowsum[N128];
    __shared__ int   sqCount;

    const int tid  = threadIdx.x;
    const int lane = tid & 31;
    const int wave = tid >> 5;
    const size_t base = (size_t)blockIdx.x * (N128 * N128);

    // ---- load A (row-major) into padded LDS, float4 per transfer ----
#pragma unroll
    for (int i = 0; i < 16; ++i) {
        const int f   = tid + i * 256;   // float4 index 0..4095
        const int row = f >> 5;
        const int c4  = f & 31;
        v4f v = *(const v4f*)(X + base + row * N128 + c4 * 4);
        *(v4f*)&bufA[row * LDSTR + c4 * 4] = v;
    }
    __syncthreads();

    // ---- inf-norm = max abs row sum; pick s so ||A/2^s|| <= 1 ----
    if (tid < N128) {
        float s = 0.0f;
        const float* rp = &bufA[tid * LDSTR];
        for (int j = 0; j < N128; ++j) s += fabsf(rp[j]);
        rowsum[tid] = s;
    }
    __syncthreads();
    if (tid == 0) {
        float nrm = 0.0f;
        for (int j = 0; j < N128; ++j) nrm = fmaxf(nrm, rowsum[j]);
        int s = 0;
        while (nrm > 1.0f && s < 30) { nrm *= 0.5f; ++s; }
        sqCount   = s;
        rowsum[0] = ldexpf(1.0f, -s);   // 2^-s
    }
    __syncthreads();
    const float scale = rowsum[0];
    const int   s     = sqCount;        // uniform across the block

    // ---- As = A * 2^-s ;  R = I + As/12 (Taylor Horner seed, m = 12) ----
#pragma unroll
    for (int i = 0; i < 64; ++i) {
        const int e   = tid + i * 256;
        const int row = e >> 7;
        const int col = e & 127;
        const float a = bufA[row * LDSTR + col] * scale;
        bufA[row * LDSTR + col] = a;
        bufR[row * LDSTR + col] = a * (1.0f / 12.0f) + (row == col ? 1.0f : 0.0f);
    }

    // ---- Horner: R <- I + (As * R)/k, k = 11..1  (11 WMMA matmuls) ----
    float* R = bufR;
    float* T = bufT;
    for (int k = 11; k >= 1; --k) {
        mm128(bufA, R, T, 1.0f / (float)k, true, lane, wave);
        float* tmp = R; R = T; T = tmp;
    }

    // ---- squaring: R <- R*R, s times (uniform branch -> EXEC all-1s) ----
    for (int q = 0; q < s; ++q) {
        mm128(R, R, T, 1.0f, false, lane, wave);
        float* tmp = R; R = T; T = tmp;
    }
    __syncthreads();

    // ---- store result ----
#pragma unroll
    for (int i = 0; i < 16; ++i) {
        const int f   = tid + i * 256;
        const int row = f >> 5;
        const int c4  = f & 31;
        *(v4f*)(Y + base + row * N128 + c4 * 4) = *(const v4f*)&R[row * LDSTR + c4 * 4];
    }
}

extern "C" void kernel_launch(void* const* d_in, const int* in_sizes, int n_in,
                              void* d_out, int out_size, void* d_ws, size_t ws_size,
                              hipStream_t stream) {
    (void)n_in; (void)d_ws; (void)ws_size; (void)out_size;
    const float* X = (const float*)d_in[0];
    float* Y = (float*)d_out;
    const int nmat = in_sizes[0] / (N128 * N128);   // 2048

    // Allow > 64 KB dynamic LDS (198 KB per workgroup; WGP has 320 KB).
    hipFuncSetAttribute((const void*)MatrixExponential_expm_kernel,
                        hipFuncAttributeMaxDynamicSharedMemorySize,
                        (int)LDS_BYTES);

    MatrixExponential_expm_kernel<<<nmat, 256, LDS_BYTES, stream>>>(X, Y);
}